// GINNet_41360535060554
// MI455X (gfx1250) — compile-verified
//
#include <hip/hip_runtime.h>
#include <hip/hip_bf16.h>
#include <math.h>

// ---------------------------------------------------------------------------
// GIN 2-layer forward on gfx1250 (MI455X), fp32 end-to-end.
//   aggr1 = x;  aggr1[dst] += x[src]           (atomic fp32 scatter-add)
//   h1    = relu(aggr1 @ W1 + b1)              (V_WMMA_F32_16X16X4_F32)
//   aggr2 = h1; aggr2[dst] += h1[src]
//   out   = log_softmax(aggr2 @ W2 + b2)       (WMMA + in-wave softmax)
// ---------------------------------------------------------------------------

#define N_NODES 100000
#define N_EDGES 1600000
#define IN_DIM 64
#define HID_DIM 128
#define OUT_DIM 40

typedef __attribute__((ext_vector_type(2))) float v2f;
typedef __attribute__((ext_vector_type(4))) float v4f;
typedef __attribute__((ext_vector_type(8))) float v8f;

// ---- init: aggr = h (fused GIN self term), vectorized 16B copies ----------
__global__ __launch_bounds__(256) void gin_copy_f4(const v4f* __restrict__ src,
                                                   v4f* __restrict__ dst) {
    unsigned i = blockIdx.x * 256u + threadIdx.x;
    dst[i] = src[i];
}

// ---- edge scatter-add, layer 1: 64 feats/edge, thread = (edge, feat) ------
__global__ __launch_bounds__(256) void gin_scatter64(
    const float* __restrict__ h, const long long* __restrict__ srcIdx,
    const long long* __restrict__ dstIdx, float* __restrict__ aggr) {
    unsigned gid = blockIdx.x * 256u + threadIdx.x;   // < N_EDGES*64
    unsigned e = gid >> 6;
    unsigned f = gid & 63u;
    int s = (int)srcIdx[e];
    int d = (int)dstIdx[e];
    atomicAdd(&aggr[(size_t)d * IN_DIM + f], h[(size_t)s * IN_DIM + f]);
}

// ---- edge scatter-add, layer 2: 128 feats/edge ----------------------------
__global__ __launch_bounds__(256) void gin_scatter128(
    const float* __restrict__ h, const long long* __restrict__ srcIdx,
    const long long* __restrict__ dstIdx, float* __restrict__ aggr) {
    unsigned gid = blockIdx.x * 256u + threadIdx.x;   // < N_EDGES*128
    unsigned e = gid >> 7;
    unsigned f = gid & 127u;
    int s = (int)srcIdx[e];
    int d = (int)dstIdx[e];
    atomicAdd(&aggr[(size_t)d * HID_DIM + f], h[(size_t)s * HID_DIM + f]);
}

// ---- GEMM1: (100000x64) @ (64x128) + bias, ReLU ---------------------------
// block = 256 threads = 8 waves; blockIdx.x = 16-row tile; wave = 16-col tile.
// Per ISA 7.12.2 f32 layouts: lane L -> M=L%16, K pair base = 2*(L/16).
__global__ __launch_bounds__(256) void gin_gemm1_relu(
    const float* __restrict__ xsum, const float* __restrict__ W1,
    const float* __restrict__ b1, float* __restrict__ h1) {
    const int lane = threadIdx.x & 31;
    const int wave = threadIdx.x >> 5;            // col tile 0..7
    const int row0 = blockIdx.x * 16;
    const int m = lane & 15;
    const int khalf = (lane >> 4) << 1;           // 0 or 2
    const int col = wave * 16 + m;

    const float* __restrict__ arow = xsum + (size_t)(row0 + m) * IN_DIM;
    v8f c = {};
#pragma unroll
    for (int k0 = 0; k0 < IN_DIM; k0 += 4) {
        const int kb = k0 + khalf;
        v2f a;
        a.x = arow[kb];
        a.y = arow[kb + 1];
        v2f b;
        b.x = W1[(size_t)kb * HID_DIM + col];
        b.y = W1[(size_t)(kb + 1) * HID_DIM + col];
        c = __builtin_amdgcn_wmma_f32_16x16x4_f32(false, a, false, b,
                                                  (short)0, c, false, false);
    }
    const float bias = b1[col];
    const int rbase = row0 + ((lane >> 4) << 3);
#pragma unroll
    for (int r = 0; r < 8; ++r) {
        float v = c[r] + bias;
        v = v > 0.f ? v : 0.f;
        h1[(size_t)(rbase + r) * HID_DIM + col] = v;
    }
}

// ---- GEMM2 + log_softmax: (100000x128) @ (128x40) -------------------------
// One wave owns a full 16-row x 48-col tile (40 real cols + 8 zero-padded),
// so softmax reduces within 16-lane halves via shuffles — no LDS round-trip.
__global__ __launch_bounds__(256) void gin_gemm2_lsm(
    const float* __restrict__ hsum, const float* __restrict__ W2,
    const float* __restrict__ b2, float* __restrict__ out) {
    const int lane = threadIdx.x & 31;
    const int wave = threadIdx.x >> 5;
    const int tile = blockIdx.x * 8 + wave;       // wave-uniform
    if (tile >= N_NODES / 16) return;             // EXEC stays all-ones below

    const int row0 = tile * 16;
    const int m = lane & 15;
    const int khalf = (lane >> 4) << 1;
    const int col0 = m, col1 = 16 + m, col2 = 32 + m;
    const bool valid2 = (m < 8);                  // col2 < 40

    const float* __restrict__ arow = hsum + (size_t)(row0 + m) * HID_DIM;
    v8f c0 = {}, c1 = {}, c2 = {};
#pragma unroll 4
    for (int k0 = 0; k0 < HID_DIM; k0 += 4) {
        const int kb = k0 + khalf;
        v2f a;
        a.x = arow[kb];
        a.y = arow[kb + 1];
        v2f b0, b1v, b2v;
        b0.x = W2[(size_t)kb * OUT_DIM + col0];
        b0.y = W2[(size_t)(kb + 1) * OUT_DIM + col0];
        b1v.x = W2[(size_t)kb * OUT_DIM + col1];
        b1v.y = W2[(size_t)(kb + 1) * OUT_DIM + col1];
        b2v.x = valid2 ? W2[(size_t)kb * OUT_DIM + col2] : 0.f;
        b2v.y = valid2 ? W2[(size_t)(kb + 1) * OUT_DIM + col2] : 0.f;
        c0 = __builtin_amdgcn_wmma_f32_16x16x4_f32(false, a, false, b0,
                                                   (short)0, c0, false, false);
        c1 = __builtin_amdgcn_wmma_f32_16x16x4_f32(false, a, false, b1v,
                                                   (short)0, c1, false, false);
        c2 = __builtin_amdgcn_wmma_f32_16x16x4_f32(false, a, false, b2v,
                                                   (short)0, c2, false, false);
    }

    const float bias0 = b2[col0];
    const float bias1 = b2[col1];
    const float bias2 = valid2 ? b2[col2] : 0.f;
    const int rbase = row0 + ((lane >> 4) << 3);

#pragma unroll
    for (int r = 0; r < 8; ++r) {
        const float v0 = c0[r] + bias0;
        const float v1 = c1[r] + bias1;
        const float v2 = valid2 ? (c2[r] + bias2) : -INFINITY;
        // row-wise max over 16-lane half (lanes of one row stay in same half)
        float mx = fmaxf(fmaxf(v0, v1), v2);
#pragma unroll
        for (int off = 1; off < 16; off <<= 1)
            mx = fmaxf(mx, __shfl_xor(mx, off, 32));
        float s = __expf(v0 - mx) + __expf(v1 - mx) +
                  (valid2 ? __expf(v2 - mx) : 0.f);
#pragma unroll
        for (int off = 1; off < 16; off <<= 1)
            s += __shfl_xor(s, off, 32);
        const float lse = mx + __logf(s);
        const size_t rowoff = (size_t)(rbase + r) * OUT_DIM;
        out[rowoff + col0] = v0 - lse;
        out[rowoff + col1] = v1 - lse;
        if (valid2) out[rowoff + col2] = v2 - lse;
    }
}

extern "C" void kernel_launch(void* const* d_in, const int* in_sizes, int n_in,
                              void* d_out, int out_size, void* d_ws, size_t ws_size,
                              hipStream_t stream) {
    const float* x = (const float*)d_in[0];
    const long long* ei = (const long long*)d_in[1];   // int64 [2, E]
    const float* W1 = (const float*)d_in[2];
    const float* b1 = (const float*)d_in[3];
    const float* W2 = (const float*)d_in[4];
    const float* b2 = (const float*)d_in[5];
    float* out = (float*)d_out;

    const long long* srcIdx = ei;
    const long long* dstIdx = ei + N_EDGES;

    // workspace layout (fp32): aggr1 | h1 | aggr2
    float* aggr1 = (float*)d_ws;                               // 100000*64
    float* h1 = aggr1 + (size_t)N_NODES * IN_DIM;              // 100000*128
    float* aggr2 = h1 + (size_t)N_NODES * HID_DIM;             // 100000*128

    // layer 1
    gin_copy_f4<<<(N_NODES * IN_DIM) / (4 * 256), 256, 0, stream>>>(
        (const v4f*)x, (v4f*)aggr1);
    gin_scatter64<<<(N_EDGES * 64) / 256, 256, 0, stream>>>(x, srcIdx, dstIdx,
                                                            aggr1);
    gin_gemm1_relu<<<N_NODES / 16, 256, 0, stream>>>(aggr1, W1, b1, h1);

    // layer 2
    gin_copy_f4<<<(N_NODES * HID_DIM) / (4 * 256), 256, 0, stream>>>(
        (const v4f*)h1, (v4f*)aggr2);
    gin_scatter128<<<(N_EDGES * 128) / 256, 256, 0, stream>>>(h1, srcIdx,
                                                              dstIdx, aggr2);
    gin_gemm2_lsm<<<(N_NODES / 16 + 7) / 8, 256, 0, stream>>>(aggr2, W2, b2,
                                                              out);
}